// DialogueRNNCell_8899172238013
// MI455X (gfx1250) — compile-verified
//
#include <hip/hip_runtime.h>
#include <hip/hip_bf16.h>
#include <math.h>
#include <stdint.h>

// ---------------------------------------------------------------------------
// DialogueRNN cell forward for MI455X (gfx1250, wave32).
// fp32 throughout (matches reference); GEMMs via V_WMMA_F32_16X16X4_F32 with
// async global->LDS staging (GLOBAL_LOAD_ASYNC_TO_LDS_B128 + ASYNCcnt).
// ---------------------------------------------------------------------------

#define BB 2048          // batch
#define TT 256           // history length
#define PP 2             // parties
#define DG 256           // D_g == D_p == D_e == GRU hidden
#define DM 512           // D_m
#define H3 768           // 3 * hidden
#define BT (BB * TT)

typedef __attribute__((ext_vector_type(2))) float v2f;
typedef __attribute__((ext_vector_type(8))) float v8f;

__device__ __forceinline__ float sigmoidf_(float x) { return 1.0f / (1.0f + __expf(-x)); }

__device__ __forceinline__ float wave_sum(float v) {
#pragma unroll
  for (int off = 16; off > 0; off >>= 1) v += __shfl_xor(v, off, 32);
  return v;
}
__device__ __forceinline__ float wave_max(float v) {
#pragma unroll
  for (int off = 16; off > 0; off >>= 1) v = fmaxf(v, __shfl_xor(v, off, 32));
  return v;
}
__device__ __forceinline__ float dot4(float4 a, float4 b) {
  return a.x * b.x + a.y * b.y + a.z * b.z + a.w * b.w;
}
__device__ __forceinline__ float4 add4(float4 a, float4 b) {
  float4 r; r.x = a.x + b.x; r.y = a.y + b.y; r.z = a.z + b.z; r.w = a.w + b.w; return r;
}
__device__ __forceinline__ float4 scale4(float s, float4 v) {
  float4 r; r.x = s * v.x; r.y = s * v.y; r.z = s * v.z; r.w = s * v.w; return r;
}

// async 16B global -> LDS copy per lane (VGLOBAL GLOBAL_LOAD_ASYNC_TO_LDS_B128,
// GV addressing; tracked by ASYNCcnt). lds_off = wave-relative LDS byte addr
// (low 32 bits of the flat shared-memory address per the aperture mapping).
__device__ __forceinline__ void async_copy_b128(uint32_t lds_off, const float* g) {
  asm volatile("global_load_async_to_lds_b128 %0, %1, off"
               :: "v"(lds_off), "v"(g)
               : "memory");
}
__device__ __forceinline__ uint32_t lds_addr(const void* p) {
  return (uint32_t)(uintptr_t)p;
}

// ---------------- WMMA fp32 GEMM:  C[M,N] = A[M,K] @ W[N,K]^T + bias[N] -----
// block = 256 threads = 8 waves; macro tile 64(M) x 32(N); K panel of 32.
// LDS double-buffered; panel k+1 streamed with async-to-LDS b128 copies while
// the 8 WMMAs of panel k run; one s_wait_asynccnt + barrier per panel.
#define GEMM_BM 64
#define GEMM_BN 32
#define GEMM_KC 32
#define LDS_S 36   // row stride in floats (16B aligned, bank-rotating pad)

__global__ __launch_bounds__(256) void gemm_bias_kernel(
    const float* __restrict__ A, const float* __restrict__ W,
    const float* __restrict__ bias, float* __restrict__ C, int M, int N, int K) {
  __shared__ float As[2][GEMM_BM * LDS_S];
  __shared__ float Ws[2][GEMM_BN * LDS_S];
  const int tid = threadIdx.x;
  const int lane = tid & 31;
  const int wv = tid >> 5;
  const int wr = wv & 3;             // wave M sub-tile (0..3)
  const int wc = wv >> 2;            // wave N sub-tile (0..1)
  const int m0 = blockIdx.y * GEMM_BM;
  const int n0 = blockIdx.x * GEMM_BN;
  const int l15 = lane & 15;
  const int kh = (lane >> 4) << 1;   // half-wave K offset: 0 or 2

  // staging coordinates: 8 float4 per tile row, one row-slot per 8 threads
  const int sr = tid >> 3;           // 0..31
  const int sc = (tid & 7) * 4;      // float offset within K-panel
  const float* Ap0 = A + (size_t)(m0 + sr) * K + sc;        // rows 0..31
  const float* Ap1 = A + (size_t)(m0 + 32 + sr) * K + sc;   // rows 32..63
  const float* Wp  = W + (size_t)(n0 + sr) * K + sc;

  uint32_t a0o[2], a1o[2], wo[2];
#pragma unroll
  for (int q = 0; q < 2; ++q) {
    a0o[q] = lds_addr(&As[q][sr * LDS_S + sc]);
    a1o[q] = lds_addr(&As[q][(32 + sr) * LDS_S + sc]);
    wo[q]  = lds_addr(&Ws[q][sr * LDS_S + sc]);
  }

  // prologue: stream panel 0 into buffer 0
  async_copy_b128(a0o[0], Ap0);
  async_copy_b128(a1o[0], Ap1);
  async_copy_b128(wo[0],  Wp);

  v8f acc = {0.f, 0.f, 0.f, 0.f, 0.f, 0.f, 0.f, 0.f};
  const int npan = K / GEMM_KC;

  for (int p = 0; p < npan; ++p) {
    const int cur = p & 1;
    if (p + 1 < npan) {  // stream next panel into the other buffer
      const int koff = (p + 1) * GEMM_KC;
      const int nxt = cur ^ 1;
      async_copy_b128(a0o[nxt], Ap0 + koff);
      async_copy_b128(a1o[nxt], Ap1 + koff);
      async_copy_b128(wo[nxt],  Wp + koff);
      asm volatile("s_wait_asynccnt 3" ::: "memory");  // panel p landed
    } else {
      asm volatile("s_wait_asynccnt 0" ::: "memory");
    }
    __syncthreads();

    const float* Arow = &As[cur][(wr * 16 + l15) * LDS_S];
    const float* Wrow = &Ws[cur][(wc * 16 + l15) * LDS_S];
    v2f af[GEMM_KC / 4], bf[GEMM_KC / 4];
#pragma unroll
    for (int kk = 0; kk < GEMM_KC / 4; ++kk) {
      af[kk] = *(const v2f*)(Arow + kk * 4 + kh);   // ds_load_b64
      bf[kk] = *(const v2f*)(Wrow + kk * 4 + kh);
    }
#pragma unroll
    for (int kk = 0; kk < GEMM_KC / 4; ++kk)
      acc = __builtin_amdgcn_wmma_f32_16x16x4_f32(false, af[kk], false, bf[kk],
                                                  0, acc, false, false);
    __syncthreads();  // all waves done reading before buffer reuse
  }
  const int col = n0 + wc * 16 + l15;
  const float bv = bias ? bias[col] : 0.0f;
  const int mb = m0 + wr * 16 + ((lane >> 4) << 3);  // C layout: v[r] -> M=r / r+8
#pragma unroll
  for (int r = 0; r < 8; ++r) C[(size_t)(mb + r) * N + col] = acc[r] + bv;
}

// ---------------- small helper kernels -------------------------------------

// sinusoidal positional-encoding table pe[t][d], t=blockIdx, d=threadIdx
__global__ void pe_kernel(float* __restrict__ pe) {
  const int t = blockIdx.x, d = threadIdx.x;
  const float dv = __expf((float)(d & ~1) * (-9.2103403719761836f / (float)DG));
  const float ang = (float)t * dv;
  pe[t * DG + d] = (d & 1) ? __cosf(ang) : __sinf(ang);
}

__global__ void argmax_kernel(const float* __restrict__ qmask, int* __restrict__ idx) {
  const int b = blockIdx.x * blockDim.x + threadIdx.x;
  if (b < BB) idx[b] = (qmask[b * PP + 1] > qmask[b * PP + 0]) ? 1 : 0;
}

// sel[b][d4] = q[b][idx[b]][d4]  (float4 elements)
__global__ void gather_party_kernel(const float* __restrict__ q,
                                    const int* __restrict__ idx,
                                    float* __restrict__ out) {
  const int i = blockIdx.x * blockDim.x + threadIdx.x;   // over BB*DG/4
  if (i >= BB * (DG / 4)) return;
  const int b = i >> 6, d4 = i & (DG / 4 - 1);
  ((float4*)out)[i] =
      ((const float4*)q)[((size_t)b * PP + idx[b]) * (DG / 4) + d4];
}

// out[b] = [ x[b] (dx) | y[b] (dy) ]  (dx, dy multiples of 4)
__global__ void concat2_kernel(const float* __restrict__ x, int dx,
                               const float* __restrict__ y, int dy,
                               float* __restrict__ out, int rows) {
  const int D4 = (dx + dy) / 4;
  const int i = blockIdx.x * blockDim.x + threadIdx.x;
  if (i >= rows * D4) return;
  const int r = i / D4, j4 = i - r * D4;
  ((float4*)out)[i] = (j4 < dx / 4)
                          ? ((const float4*)x)[(size_t)r * (dx / 4) + j4]
                          : ((const float4*)y)[(size_t)r * (dy / 4) + (j4 - dx / 4)];
}

#define GRU_COMP(c)                                            \
  {                                                            \
    float rr = sigmoidf_(gir.c + ghr.c);                       \
    float zz = sigmoidf_(giz.c + ghz.c);                       \
    float nn = tanhf(gin.c + rr * ghn.c);                      \
    o.c = (1.0f - zz) * nn + zz * hv.c;                        \
  }

// standard GRU combine: h' = (1-z)*n + z*h   (gate order r,z,n; biases in gi/gh)
__global__ void gru_combine_kernel(const float* __restrict__ gi,
                                   const float* __restrict__ gh,
                                   const float* __restrict__ h,
                                   float* __restrict__ out, int rows) {
  const int i = blockIdx.x * blockDim.x + threadIdx.x;   // over rows*DG/4
  if (i >= rows * (DG / 4)) return;
  const int b = i >> 6, j = (i & (DG / 4 - 1)) * 4;
  const float* gib = gi + (size_t)b * H3;
  const float* ghb = gh + (size_t)b * H3;
  float4 gir = *(const float4*)&gib[j],          ghr = *(const float4*)&ghb[j];
  float4 giz = *(const float4*)&gib[DG + j],     ghz = *(const float4*)&ghb[DG + j];
  float4 gin = *(const float4*)&gib[2 * DG + j], ghn = *(const float4*)&ghb[2 * DG + j];
  float4 hv = ((const float4*)h)[i];
  float4 o;
  GRU_COMP(x) GRU_COMP(y) GRU_COMP(z) GRU_COMP(w)
  ((float4*)out)[i] = o;
}

#define PGRU_COMP(c)                                           \
  {                                                            \
    float rr = sigmoidf_(gir.c + ghr.c);                       \
    float zz = sigmoidf_(giz.c + ghz.c);                       \
    float nn = tanhf(gin.c + rr * ghn.c);                      \
    float qs = (1.0f - zz) * nn + zz * hv.c;                   \
    o.c = hv.c + m * (qs - hv.c);                              \
  }

// party GRU: gi shared per batch row (broadcast input), masked blend with q0
__global__ void party_combine_kernel(const float* __restrict__ gi,
                                     const float* __restrict__ gh,
                                     const float* __restrict__ q0,
                                     const float* __restrict__ qmask,
                                     float* __restrict__ qout) {
  const int i = blockIdx.x * blockDim.x + threadIdx.x;   // over BB*PP*DG/4
  if (i >= BB * PP * (DG / 4)) return;
  const int bp = i >> 6, j = (i & (DG / 4 - 1)) * 4;
  const int b = bp >> 1, p = bp & 1;
  const float* gib = gi + (size_t)b * H3;
  const float* ghb = gh + (size_t)bp * H3;
  float4 gir = *(const float4*)&gib[j],          ghr = *(const float4*)&ghb[j];
  float4 giz = *(const float4*)&gib[DG + j],     ghz = *(const float4*)&ghb[DG + j];
  float4 gin = *(const float4*)&gib[2 * DG + j], ghn = *(const float4*)&ghb[2 * DG + j];
  float4 hv = ((const float4*)q0)[i];
  const float m = qmask[b * PP + p];
  float4 o;
  PGRU_COMP(x) PGRU_COMP(y) PGRU_COMP(z) PGRU_COMP(w)
  ((float4*)qout)[i] = o;
}

// scores for two heads over one (T,B,D) tensor; one wave per (t,b) row.
// s[h][b*T + t] = dot(M[t,b,:] + pe[t,:], w_h)
__global__ __launch_bounds__(256) void att_score_kernel(
    const float* __restrict__ Mten, const float* __restrict__ pe,
    const float* __restrict__ w1, const float* __restrict__ w2,
    float* __restrict__ s1, float* __restrict__ s2) {
  const int wid = blockIdx.x * 8 + (threadIdx.x >> 5);  // == t*B + b
  const int lane = threadIdx.x & 31;
  const int t = wid / BB, b = wid - t * BB;
  const float4* row = (const float4*)(Mten + (size_t)wid * DG);
  const float4* per = (const float4*)(pe + (size_t)t * DG);
  const float4* w1v = (const float4*)w1;
  const float4* w2v = (const float4*)w2;
  float a1 = 0.f, a2 = 0.f;
#pragma unroll
  for (int j = 0; j < DG / 128; ++j) {           // 2 x b128 per lane
    const int q = lane + 32 * j;
    const float4 v = add4(row[q], per[q]);
    a1 += dot4(v, w1v[q]);
    a2 += dot4(v, w2v[q]);
  }
  a1 = wave_sum(a1);
  a2 = wave_sum(a2);
  if (lane == 0) { s1[b * TT + t] = a1; s2[b * TT + t] = a2; }
}

// softmax over T for 4 heads stacked: rows = 4*B, row length T; one wave/row
__global__ __launch_bounds__(256) void softmax_T_kernel(
    const float* __restrict__ s, float* __restrict__ al) {
  const int wid = blockIdx.x * 8 + (threadIdx.x >> 5);
  const int lane = threadIdx.x & 31;
  const float4* row = (const float4*)(s + (size_t)wid * TT);
  float4 v[TT / 128];
  float mx = -3.0e38f;
#pragma unroll
  for (int j = 0; j < TT / 128; ++j) {
    v[j] = row[lane + 32 * j];
    mx = fmaxf(mx, fmaxf(fmaxf(v[j].x, v[j].y), fmaxf(v[j].z, v[j].w)));
  }
  mx = wave_max(mx);
  float sum = 0.f;
#pragma unroll
  for (int j = 0; j < TT / 128; ++j) {
    v[j].x = __expf(v[j].x - mx); v[j].y = __expf(v[j].y - mx);
    v[j].z = __expf(v[j].z - mx); v[j].w = __expf(v[j].w - mx);
    sum += v[j].x + v[j].y + v[j].z + v[j].w;
  }
  sum = wave_sum(sum);
  const float inv = 1.0f / sum;
  float4* orow = (float4*)(al + (size_t)wid * TT);
#pragma unroll
  for (int j = 0; j < TT / 128; ++j) orow[lane + 32 * j] = scale4(inv, v[j]);
}

// alpha output: sum of the 4 head alphas, (B,1,T) layout
__global__ void alpha_sum_kernel(const float* __restrict__ al, float* __restrict__ out) {
  const int i = blockIdx.x * blockDim.x + threadIdx.x;   // over BT/4
  if (i >= BT / 4) return;
  const float4* a4 = (const float4*)al;
  ((float4*)out)[i] = add4(add4(a4[i], a4[BT / 4 + i]),
                           add4(a4[2 * (BT / 4) + i], a4[3 * (BT / 4) + i]));
}

// context: c_h[b,d] = sum_t alpha_h[b,t] * (M[t,b,d] + pe[t,d]); block per b
__global__ __launch_bounds__(256) void context_kernel(
    const float* __restrict__ Mten, const float* __restrict__ pe,
    const float* __restrict__ al1, const float* __restrict__ al2,
    float* __restrict__ c1, float* __restrict__ c2) {
  const int b = blockIdx.x, d = threadIdx.x;
  float s1 = 0.f, s2 = 0.f;
  for (int t = 0; t < TT; ++t) {
    if (t + 8 < TT)
      __builtin_prefetch(&Mten[((size_t)(t + 8) * BB + b) * DG + d], 0, 0);
    const float v = Mten[((size_t)t * BB + b) * DG + d] + pe[t * DG + d];
    s1 += al1[b * TT + t] * v;
    s2 += al2[b * TT + t] * v;
  }
  c1[(size_t)b * DG + d] = s1;
  c2[(size_t)b * DG + d] = s2;
}

// self-attention ('general2', N=2): out[b] = [a0*c1[b], a1*c2[b]]; wave per b
__global__ __launch_bounds__(256) void self_att_kernel(
    const float* __restrict__ c1, const float* __restrict__ c2,
    const float* __restrict__ w, const float* __restrict__ bias,
    float* __restrict__ out) {
  const int b = blockIdx.x * 8 + (threadIdx.x >> 5);
  const int lane = threadIdx.x & 31;
  const float4* r1 = (const float4*)(c1 + (size_t)b * DG);
  const float4* r2 = (const float4*)(c2 + (size_t)b * DG);
  const float4* wv = (const float4*)w;
  float s1 = 0.f, s2 = 0.f;
#pragma unroll
  for (int j = 0; j < DG / 128; ++j) {
    const int q = lane + 32 * j;
    s1 += dot4(r1[q], wv[q]);
    s2 += dot4(r2[q], wv[q]);
  }
  s1 = wave_sum(s1) + bias[0];
  s2 = wave_sum(s2) + bias[0];
  const float mx = fmaxf(s1, s2);
  const float e1 = __expf(s1 - mx), e2 = __expf(s2 - mx);
  const float inv = 1.0f / (e1 + e2);
  const float a1 = e1 * inv, a2 = e2 * inv;
  float4* ob = (float4*)(out + (size_t)b * 2 * DG);
#pragma unroll
  for (int j = 0; j < DG / 128; ++j) {
    const int q = lane + 32 * j;
    ob[q] = scale4(a1, r1[q]);
    ob[DG / 4 + q] = scale4(a2, r2[q]);
  }
}

// ---------------------------------------------------------------------------
extern "C" void kernel_launch(void* const* d_in, const int* in_sizes, int n_in,
                              void* d_out, int out_size, void* d_ws, size_t ws_size,
                              hipStream_t stream) {
  // inputs (setup_inputs insertion order)
  const float* Ua      = (const float*)d_in[0];
  const float* Ub      = (const float*)d_in[1];
  const float* qmask   = (const float*)d_in[2];
  const float* ghist_a = (const float*)d_in[3];
  const float* ghist_b = (const float*)d_in[4];
  const float* q0_a    = (const float*)d_in[5];
  const float* q0_b    = (const float*)d_in[6];
  const float* e0_a    = (const float*)d_in[7];
  const float* e0_b    = (const float*)d_in[8];
  // params leaves, insertion order
  const float* Wda = (const float*)d_in[9],  *bda = (const float*)d_in[10];
  const float* Wdb = (const float*)d_in[11], *bdb = (const float*)d_in[12];
  const float* gWa_ih = (const float*)d_in[13], *gWa_hh = (const float*)d_in[14];
  const float* gba_ih = (const float*)d_in[15], *gba_hh = (const float*)d_in[16];
  const float* gWb_ih = (const float*)d_in[17], *gWb_hh = (const float*)d_in[18];
  const float* gbb_ih = (const float*)d_in[19], *gbb_hh = (const float*)d_in[20];
  const float* pWa_ih = (const float*)d_in[21], *pWa_hh = (const float*)d_in[22];
  const float* pba_ih = (const float*)d_in[23], *pba_hh = (const float*)d_in[24];
  const float* pWb_ih = (const float*)d_in[25], *pWb_hh = (const float*)d_in[26];
  const float* pbb_ih = (const float*)d_in[27], *pbb_hh = (const float*)d_in[28];
  const float* eWa_ih = (const float*)d_in[29], *eWa_hh = (const float*)d_in[30];
  const float* eba_ih = (const float*)d_in[31], *eba_hh = (const float*)d_in[32];
  const float* eWb_ih = (const float*)d_in[33], *eWb_hh = (const float*)d_in[34];
  const float* ebb_ih = (const float*)d_in[35], *ebb_hh = (const float*)d_in[36];
  const float* w_att1 = (const float*)d_in[37];
  const float* w_att2 = (const float*)d_in[38];
  const float* w_att3 = (const float*)d_in[39];
  const float* w_att4 = (const float*)d_in[40];
  const float* w_sa1  = (const float*)d_in[41], *b_sa1 = (const float*)d_in[42];
  const float* w_sa2  = (const float*)d_in[43], *b_sa2 = (const float*)d_in[44];
  const float* Wd1 = (const float*)d_in[45], *bd1 = (const float*)d_in[46];
  const float* Wd2 = (const float*)d_in[47], *bd2 = (const float*)d_in[48];
  (void)in_sizes; (void)n_in; (void)out_size; (void)ws_size;

  float* out = (float*)d_out;
  float* out_ga = out + 0;
  float* out_qa = out + 524288;
  float* out_ea = out + 1572864;
  float* out_gb = out + 2097152;
  float* out_qb = out + 2621440;
  float* out_eb = out + 3670016;
  float* out_al = out + 4194304;

  // workspace layout (floats)
  float* ws = (float*)d_ws;
  size_t o = 0;
  float* XA  = ws + o; o += (size_t)BB * DM;        // densed Ua
  float* XB  = ws + o; o += (size_t)BB * DM;        // densed Ub
  float* PE  = ws + o; o += (size_t)TT * DG;        // pos-enc table
  float* SC  = ws + o; o += 4ull * BT;              // 4 heads of raw scores (B,T)
  float* AL  = ws + o; o += 4ull * BT;              // 4 heads of alphas (B,T)
  float* CAA = ws + o; o += (size_t)BB * DG;
  float* CBA = ws + o; o += (size_t)BB * DG;
  float* CAB = ws + o; o += (size_t)BB * DG;
  float* CBB = ws + o; o += (size_t)BB * DG;
  float* SAA = ws + o; o += (size_t)BB * 2 * DG;    // weighted self-att stack a
  float* SAB = ws + o; o += (size_t)BB * 2 * DG;    // weighted self-att stack b
  float* CA  = ws + o; o += (size_t)BB * DG;        // c_a after dense1
  float* CB  = ws + o; o += (size_t)BB * DG;        // c_b after dense2
  float* CAT = ws + o; o += (size_t)BB * H3;        // concat scratch (reused)
  float* GI  = ws + o; o += (size_t)BB * H3;        // input-gate GEMM out (reused)
  float* GH  = ws + o; o += (size_t)BB * PP * H3;   // hidden-gate GEMM out (reused)
  float* SEL = ws + o; o += (size_t)BB * DG;        // party gather (reused)
  int* IDX = (int*)(ws + o); o += BB;

  auto gemm = [&](const float* A, const float* W, const float* bias, float* C,
                  int M, int N, int K) {
    dim3 grid(N / GEMM_BN, M / GEMM_BM);
    gemm_bias_kernel<<<grid, 256, 0, stream>>>(A, W, bias, C, M, N, K);
  };

  // ---- prelude -------------------------------------------------------------
  pe_kernel<<<TT, DG, 0, stream>>>(PE);
  argmax_kernel<<<BB / 256, 256, 0, stream>>>(qmask, IDX);
  gemm(Ua, Wda, bda, XA, BB, DM, 1024);             // dense_a
  gemm(Ub, Wdb, bdb, XB, BB, DM, 512);              // dense_b

  const float* ha_last = ghist_a + (size_t)(TT - 1) * BB * DG;
  const float* hb_last = ghist_b + (size_t)(TT - 1) * BB * DG;

  // ---- global GRU a --------------------------------------------------------
  gather_party_kernel<<<(BB * DG / 4) / 256, 256, 0, stream>>>(q0_a, IDX, SEL);
  concat2_kernel<<<(BB * H3 / 4) / 256, 256, 0, stream>>>(XA, DM, SEL, DG, CAT, BB);
  gemm(CAT, gWa_ih, gba_ih, GI, BB, H3, H3);
  gemm(ha_last, gWa_hh, gba_hh, GH, BB, H3, DG);
  gru_combine_kernel<<<(BB * DG / 4) / 256, 256, 0, stream>>>(GI, GH, ha_last, out_ga, BB);

  // ---- global GRU b --------------------------------------------------------
  gather_party_kernel<<<(BB * DG / 4) / 256, 256, 0, stream>>>(q0_b, IDX, SEL);
  concat2_kernel<<<(BB * H3 / 4) / 256, 256, 0, stream>>>(XB, DM, SEL, DG, CAT, BB);
  gemm(CAT, gWb_ih, gbb_ih, GI, BB, H3, H3);
  gemm(hb_last, gWb_hh, gbb_hh, GH, BB, H3, DG);
  gru_combine_kernel<<<(BB * DG / 4) / 256, 256, 0, stream>>>(GI, GH, hb_last, out_gb, BB);

  // ---- time attention (bandwidth-bound streaming passes) -------------------
  att_score_kernel<<<(TT * BB) / 8, 256, 0, stream>>>(ghist_a, PE, w_att1, w_att2,
                                                      SC + 0, SC + BT);
  att_score_kernel<<<(TT * BB) / 8, 256, 0, stream>>>(ghist_b, PE, w_att3, w_att4,
                                                      SC + 2 * BT, SC + 3 * BT);
  softmax_T_kernel<<<(4 * BB) / 8, 256, 0, stream>>>(SC, AL);
  alpha_sum_kernel<<<(BT / 4) / 256, 256, 0, stream>>>(AL, out_al);
  context_kernel<<<BB, DG, 0, stream>>>(ghist_a, PE, AL + 0, AL + BT, CAA, CBA);
  context_kernel<<<BB, DG, 0, stream>>>(ghist_b, PE, AL + 2 * BT, AL + 3 * BT, CAB, CBB);

  // ---- self-attention + output denses -------------------------------------
  self_att_kernel<<<BB / 8, 256, 0, stream>>>(CAA, CAB, w_sa1, b_sa1, SAA);
  gemm(SAA, Wd1, bd1, CA, BB, DG, 2 * DG);
  self_att_kernel<<<BB / 8, 256, 0, stream>>>(CBB, CBA, w_sa2, b_sa2, SAB);
  gemm(SAB, Wd2, bd2, CB, BB, DG, 2 * DG);

  // ---- party + emotion update a -------------------------------------------
  concat2_kernel<<<(BB * H3 / 4) / 256, 256, 0, stream>>>(XA, DM, CA, DG, CAT, BB);
  gemm(CAT, pWa_ih, pba_ih, GI, BB, H3, H3);              // gi shared over parties
  gemm(q0_a, pWa_hh, pba_hh, GH, BB * PP, H3, DG);
  party_combine_kernel<<<(BB * PP * DG / 4) / 256, 256, 0, stream>>>(GI, GH, q0_a, qmask, out_qa);
  gather_party_kernel<<<(BB * DG / 4) / 256, 256, 0, stream>>>(out_qa, IDX, SEL);
  gemm(SEL, eWa_ih, eba_ih, GI, BB, H3, DG);
  gemm(e0_a, eWa_hh, eba_hh, GH, BB, H3, DG);
  gru_combine_kernel<<<(BB * DG / 4) / 256, 256, 0, stream>>>(GI, GH, e0_a, out_ea, BB);

  // ---- party + emotion update b -------------------------------------------
  concat2_kernel<<<(BB * H3 / 4) / 256, 256, 0, stream>>>(XB, DM, CB, DG, CAT, BB);
  gemm(CAT, pWb_ih, pbb_ih, GI, BB, H3, H3);
  gemm(q0_b, pWb_hh, pbb_hh, GH, BB * PP, H3, DG);
  party_combine_kernel<<<(BB * PP * DG / 4) / 256, 256, 0, stream>>>(GI, GH, q0_b, qmask, out_qb);
  gather_party_kernel<<<(BB * DG / 4) / 256, 256, 0, stream>>>(out_qb, IDX, SEL);
  gemm(SEL, eWb_ih, ebb_ih, GI, BB, H3, DG);
  gemm(e0_b, eWb_hh, ebb_hh, GH, BB, H3, DG);
  gru_combine_kernel<<<(BB * DG / 4) / 256, 256, 0, stream>>>(GI, GH, e0_b, out_eb, BB);
}